// ChamferDistanceLoss_60026462929423
// MI455X (gfx1250) — compile-verified
//
#include <hip/hip_runtime.h>
#include <math.h>

// Chamfer distance, B=4, N=M=8192, D=3, fp32.
// dist^2(p,q) = |p|^2 - 2 p.q + |q|^2. We fold (|p|^2 - 2 p.q) into a single
// V_WMMA_F32_16X16X4_F32 by augmenting K=4:
//   A[M][:] = (-2px, -2py, -2pz, |p|^2),  B[:][N] = (qx, qy, qz, 1)
// C/D layout (16x16 f32): VGPR i holds M=i (lanes 0-15, N=lane) and M=i+8
// (lanes 16-31, N=lane-16)  => min over M (the P axis) is 7 in-lane mins +
// one shfl_xor(16). |q|^2 is a per-lane constant added after the min.

typedef float v2f __attribute__((ext_vector_type(2)));
typedef float v8f __attribute__((ext_vector_type(8)));

#define BATCH 4
#define NPTS 8192
#define QTILES_PER_BATCH (NPTS / 16)            // 512
#define WAVES_TOTAL (BATCH * QTILES_PER_BATCH)  // 2048 waves per direction
#define WAVES_PER_BLOCK 8
#define BLOCKS (WAVES_TOTAL / WAVES_PER_BLOCK)  // 256 blocks of 256 threads

// One wave owns a 16-point tile of Q and scans all of P in 16-row chunks.
// P plays the A (M) role, Q plays the B (N) role.
__global__ __launch_bounds__(256) void chamfer_dir_kernel(
    const float* __restrict__ P, const float* __restrict__ Q,
    float* __restrict__ partials) {
  const int lane = threadIdx.x & 31;
  const int wave = blockIdx.x * WAVES_PER_BLOCK + (threadIdx.x >> 5);
  const int b = wave / QTILES_PER_BATCH;
  const int qbase = (wave % QTILES_PER_BATCH) * 16;
  const int col = lane & 15;  // N for B/C/D, M for A
  const int hi = lane >> 4;   // 0: K=0,1   1: K=2,3

  // ---- B operand: fixed Q tile for this wave ----
  const float* qp = Q + (size_t)(b * NPTS + qbase + col) * 3;
  const float qx = qp[0], qy = qp[1], qz = qp[2];
  const float q2 = qx * qx + qy * qy + qz * qz;
  v2f bv;
  bv.x = hi ? qz : qx;
  bv.y = hi ? 1.0f : qy;

  // running min of (|p|^2 - 2 p.q) per (M-slice, N=col)
  v8f m;
#pragma unroll
  for (int i = 0; i < 8; ++i) m[i] = 3.0e38f;

  const float* pbase = P + (size_t)b * NPTS * 3 + (size_t)col * 3;
#pragma unroll 4
  for (int chunk = 0; chunk < NPTS / 16; ++chunk) {
    const float* pp = pbase + (size_t)chunk * 48;  // 16 points * 3 floats
    const float px = pp[0], py = pp[1], pz = pp[2];
    v2f av;
    av.x = hi ? (-2.0f * pz) : (-2.0f * px);
    av.y = hi ? (px * px + py * py + pz * pz) : (-2.0f * py);
    v8f c = {};
    // 8 args: (neg_a, A, neg_b, B, c_mod, C, reuse_a, reuse_b)
    c = __builtin_amdgcn_wmma_f32_16x16x4_f32(
        false, av, false, bv, (short)0, c, false, false);
    m = __builtin_elementwise_min(m, c);
  }

  // min over the 8 M-slices held in-lane, then across lane halves (M vs M+8)
  float r = m[0];
#pragma unroll
  for (int i = 1; i < 8; ++i) r = fminf(r, m[i]);
  r = fminf(r, __shfl_xor(r, 16, 32));

  float val = sqrtf(fmaxf(r + q2, 0.0f));  // min distance for point q=col

  // sum the 16 columns (lanes 16-31 hold duplicates; stop at xor 8)
  val += __shfl_xor(val, 1, 32);
  val += __shfl_xor(val, 2, 32);
  val += __shfl_xor(val, 4, 32);
  val += __shfl_xor(val, 8, 32);
  if (lane == 0) partials[wave] = val;
}

__global__ __launch_bounds__(256) void chamfer_reduce_kernel(
    const float* __restrict__ partials, float* __restrict__ out) {
  __shared__ float sm[256];
  float s = 0.0f;
  for (int i = threadIdx.x; i < 2 * WAVES_TOTAL; i += 256) s += partials[i];
  sm[threadIdx.x] = s;
  __syncthreads();
#pragma unroll
  for (int off = 128; off > 0; off >>= 1) {
    if ((int)threadIdx.x < off) sm[threadIdx.x] += sm[threadIdx.x + off];
    __syncthreads();
  }
  if (threadIdx.x == 0)
    out[0] = sm[0] / (float)(BATCH * NPTS);  // both means share denom 32768
}

extern "C" void kernel_launch(void* const* d_in, const int* in_sizes, int n_in,
                              void* d_out, int out_size, void* d_ws,
                              size_t ws_size, hipStream_t stream) {
  const float* tmpl = (const float*)d_in[0];  // [4,8192,3]
  const float* src = (const float*)d_in[1];   // [4,8192,3]
  float* partials = (float*)d_ws;             // 2*2048 floats = 16 KB
  float* out = (float*)d_out;

  // min over template per source point (min_p0_p1): P=template, Q=source
  chamfer_dir_kernel<<<BLOCKS, 256, 0, stream>>>(tmpl, src, partials);
  // min over source per template point (min_p1_p0): P=source, Q=template
  chamfer_dir_kernel<<<BLOCKS, 256, 0, stream>>>(src, tmpl,
                                                 partials + WAVES_TOTAL);
  chamfer_reduce_kernel<<<1, 256, 0, stream>>>(partials, out);
}